// LinkBertModel_61950608277772
// MI455X (gfx1250) — compile-verified
//
#include <hip/hip_runtime.h>
#include <hip/hip_bf16.h>
#include <math.h>

typedef __attribute__((ext_vector_type(2))) float v2f;
typedef __attribute__((ext_vector_type(8))) float v8f;

#define H_DIM 768
#define C_DIM 32
#define KF    800   // H + C, multiple of 16
#define D1_   512
#define D2_   512

#if __has_builtin(__builtin_amdgcn_wmma_f32_16x16x4_f32)
#define USE_WMMA_F32X4 1
#endif

// ---------------------------------------------------------------------------
// Phase 1: segment mean-pool of gathered token embeddings + concat side feats.
// One block (256 threads) per segment; segment_ids is sorted so each segment
// is a contiguous range found by binary search.
// ---------------------------------------------------------------------------
__device__ __forceinline__ int lower_bound_i32(const int* __restrict__ a, int n, int v) {
    int lo = 0, hi = n;
    while (lo < hi) {
        int mid = (lo + hi) >> 1;
        if (a[mid] < v) lo = mid + 1; else hi = mid;
    }
    return lo;
}

__global__ void __launch_bounds__(256)
pool_concat_kernel(const float* __restrict__ hs,      // [B*S, H]
                   const float* __restrict__ basec,   // [N, C]
                   const int*   __restrict__ tok,     // [T]
                   const int*   __restrict__ seg,     // [T] sorted
                   float*       __restrict__ feats,   // [N, KF]
                   int T) {
    const int n   = blockIdx.x;
    const int tid = threadIdx.x;

    const int lo = lower_bound_i32(seg, T, n);
    const int hi = lower_bound_i32(seg, T, n + 1);

    float a0 = 0.f, a1 = 0.f, a2 = 0.f;   // h = tid, tid+256, tid+512
    for (int t = lo; t < hi; ++t) {
        const float* row = hs + (size_t)tok[t] * H_DIM;
        a0 += row[tid];
        a1 += row[tid + 256];
        a2 += row[tid + 512];
    }
    const float inv = 1.0f / fmaxf((float)(hi - lo), 1.0f);
    float* f = feats + (size_t)n * KF;
    f[tid]       = a0 * inv;
    f[tid + 256] = a1 * inv;
    f[tid + 512] = a2 * inv;
    if (tid < C_DIM) f[H_DIM + tid] = basec[(size_t)n * C_DIM + tid];
}

// ---------------------------------------------------------------------------
// Phase 2: fp32 WMMA GEMM. One wave32 per 16x16 output tile, K stepped by 4
// with V_WMMA_F32_16X16X4_F32. Fused bias + optional ReLU.
//
// A-frag layout (ISA 7.12.2, 32-bit A 16x4): lanes 0-15 hold M=lane, K=k+{0,1};
// lanes 16-31 hold M=lane-16, K=k+{2,3}.
// B-frag (4x16, rows striped across lanes): lanes 0-15 -> K=k+{0,1} at N=lane;
// lanes 16-31 -> K=k+{2,3} at N=lane-16.
// C/D layout: VGPR r -> M = r (lanes 0-15) / r+8 (lanes 16-31), N = lane&15.
// ---------------------------------------------------------------------------
__global__ void __launch_bounds__(256)
wmma_gemm_bias_act(const float* __restrict__ A,     // [M, K]
                   const float* __restrict__ W,     // [K, Nout]
                   const float* __restrict__ bias,  // [Nout]
                   float*       __restrict__ out,   // [M, Nout]
                   int M, int K, int Nout,
                   int totalTiles, int tilesN, int relu) {
    const int wave = (blockIdx.x * blockDim.x + threadIdx.x) >> 5;
    if (wave >= totalTiles) return;          // whole-wave uniform exit
    const int lane  = threadIdx.x & 31;
    const int tm    = wave / tilesN;
    const int tn    = wave - tm * tilesN;
    const int lhalf = lane >> 4;             // 0 or 1
    const int l15   = lane & 15;
    const int kgrp  = lhalf << 1;            // 0 or 2

    const float* arow = A + (size_t)(tm * 16 + l15) * K;
    const float* wcol = W + (size_t)(tn * 16 + l15);

#ifdef USE_WMMA_F32X4
    v8f c = {};
    for (int k = 0; k < K; k += 4) {
        v2f a, b;
        a.x = arow[k + kgrp];
        a.y = arow[k + kgrp + 1];
        b.x = wcol[(size_t)(k + kgrp)     * Nout];
        b.y = wcol[(size_t)(k + kgrp + 1) * Nout];
        // (neg_a, A, neg_b, B, c_mod, C, reuse_a, reuse_b)
        c = __builtin_amdgcn_wmma_f32_16x16x4_f32(false, a, false, b,
                                                  (short)0, c, false, false);
    }
    const int   col   = tn * 16 + l15;
    const float bv    = bias[col];
    const int   rbase = tm * 16 + (lhalf << 3);
#pragma unroll
    for (int r = 0; r < 8; ++r) {
        float v = c[r] + bv;
        if (relu) v = fmaxf(v, 0.0f);
        out[(size_t)(rbase + r) * Nout + col] = v;
    }
#else
    // Scalar fallback with identical tile mapping (same launch geometry):
    // each lane produces the same 8 elements the WMMA layout would.
    const int   col   = tn * 16 + l15;
    const float bv    = bias[col];
    const int   rbase = tm * 16 + (lhalf << 3);
#pragma unroll
    for (int r = 0; r < 8; ++r) {
        const float* ar = A + (size_t)(rbase + r) * K;
        float acc = 0.f;
        for (int k = 0; k < K; ++k)
            acc = fmaf(ar[k], W[(size_t)k * Nout + col], acc);
        float v = acc + bv;
        if (relu) v = fmaxf(v, 0.0f);
        out[(size_t)(rbase + r) * Nout + col] = v;
    }
#endif
}

// ---------------------------------------------------------------------------
// Phase 3: head — [N,512] x [512,1] dot + bias + sigmoid. One block per row.
// ---------------------------------------------------------------------------
__global__ void __launch_bounds__(256)
head_sigmoid_kernel(const float* __restrict__ h2,   // [N, D2]
                    const float* __restrict__ w3,   // [D2]
                    const float* __restrict__ b3,   // [1]
                    float*       __restrict__ out,  // [N]
                    int D) {
    const int n   = blockIdx.x;
    const int tid = threadIdx.x;
    const float* row = h2 + (size_t)n * D;

    float s = 0.f;
    for (int k = tid; k < D; k += 256)
        s = fmaf(row[k], w3[k], s);

    // wave32 shuffle reduction
#pragma unroll
    for (int off = 16; off > 0; off >>= 1)
        s += __shfl_down(s, off, 32);

    __shared__ float red[8];
    if ((tid & 31) == 0) red[tid >> 5] = s;
    __syncthreads();
    if (tid == 0) {
        float tot = b3[0];
#pragma unroll
        for (int i = 0; i < 8; ++i) tot += red[i];
        out[n] = 1.0f / (1.0f + __expf(-tot));
    }
}

// ---------------------------------------------------------------------------
// Host launch
// ---------------------------------------------------------------------------
extern "C" void kernel_launch(void* const* d_in, const int* in_sizes, int n_in,
                              void* d_out, int out_size, void* d_ws, size_t ws_size,
                              hipStream_t stream) {
    const float* hs    = (const float*)d_in[0];  // hidden_states [B*S*H]
    const float* basec = (const float*)d_in[1];  // base_characteristics [N*C]
    const float* w1    = (const float*)d_in[2];
    const float* b1    = (const float*)d_in[3];
    const float* w2    = (const float*)d_in[4];
    const float* b2    = (const float*)d_in[5];
    const float* w3    = (const float*)d_in[6];
    const float* b3    = (const float*)d_in[7];
    const int*   tok   = (const int*)d_in[8];    // token_indices [T]
    const int*   seg   = (const int*)d_in[9];    // segment_ids [T], sorted

    const int T = in_sizes[8];
    const int N = out_size;                      // 2048 segments

    float* feats = (float*)d_ws;                        // [N, KF]
    float* h1    = feats + (size_t)N * KF;              // [N, D1]
    float* h2    = h1    + (size_t)N * D1_;             // [N, D2]

    // 1) gather + segment mean + concat
    pool_concat_kernel<<<N, 256, 0, stream>>>(hs, basec, tok, seg, feats, T);

    // 2) feats[N,800] @ w1[800,512] + b1, ReLU
    {
        const int tilesN = D1_ / 16, tilesM = N / 16;
        const int total  = tilesM * tilesN;
        const int blocks = (total * 32 + 255) / 256;
        wmma_gemm_bias_act<<<blocks, 256, 0, stream>>>(feats, w1, b1, h1,
                                                       N, KF, D1_, total, tilesN, 1);
    }
    // 3) h1[N,512] @ w2[512,512] + b2, ReLU
    {
        const int tilesN = D2_ / 16, tilesM = N / 16;
        const int total  = tilesM * tilesN;
        const int blocks = (total * 32 + 255) / 256;
        wmma_gemm_bias_act<<<blocks, 256, 0, stream>>>(h1, w2, b2, h2,
                                                       N, D1_, D2_, total, tilesN, 1);
    }
    // 4) head + sigmoid -> d_out [1, N]
    head_sigmoid_kernel<<<N, 256, 0, stream>>>(h2, w3, b3, (float*)d_out, D2_);
}